// agss_37821482009416
// MI455X (gfx1250) — compile-verified
//
#include <hip/hip_runtime.h>

typedef float v2f __attribute__((ext_vector_type(2)));
typedef float v8f __attribute__((ext_vector_type(8)));
typedef unsigned int u32x4 __attribute__((ext_vector_type(4)));
typedef int i32x4 __attribute__((ext_vector_type(4)));
typedef int i32x8 __attribute__((ext_vector_type(8)));

#define NB 32
#define NC 224
#define NT 2048
#define TT 64          // timesteps per LDS tile
#define STR 65         // LDS row stride in dwords (odd -> conflict-free banks)
#define NTILES (NT / TT)
#define F_SIGMA_MIN 0.4f
#define F_SIGMA_MAX 10.0f
#define F_SIGMA_MEAN 5.2f
#define F_BETA 0.95f
#define F_THRESH 1.0f

// Low 32 bits of a flat shared-memory pointer == LDS byte offset (ISA 10.2:
// LDS aperture translation truncates to addr[31:0]).
static __device__ __forceinline__ unsigned lds_off(const void* p) {
  return (unsigned)(unsigned long long)(size_t)p;
}

// TDM: DMA one (NC x TT) f32 tile (row stride NT elements in memory) into LDS,
// hardware-padding +1 dword after every 64-dword row -> LDS row stride = 65.
static __device__ __forceinline__ void tdm_load_tile(const float* gsrc,
                                                     unsigned lds_byte_addr) {
  unsigned long long ga = (unsigned long long)(size_t)(const void*)gsrc;
  u32x4 g0;
  g0[0] = 1u;                                   // count=1, user mode, no gather
  g0[1] = lds_byte_addr;                        // D#.lds_addr
  g0[2] = (unsigned)(ga & 0xFFFFFFFFu);         // global_addr[31:0]
  g0[3] = (unsigned)((ga >> 32) & 0x01FFFFFFu)  // global_addr[56:32]
          | 0x80000000u;                        // type=2 ("image")
  i32x8 g1;
  g1[0] = (int)((2u << 16)      // data_size = 4 bytes
              | (1u << 20)      // pad_enable
              | (5u << 22));    // pad_interval code 5 = 64 dwords (one row)
                                // pad_amount code 0 = 1 dword; mask=0 (no cluster)
  g1[1] = (int)(((unsigned)NT & 0xFFFFu) << 16);            // tensor_dim0 lo16
  g1[2] = (int)((((unsigned)NT >> 16) & 0xFFFFu)            // tensor_dim0 hi16
              | (((unsigned)NC & 0xFFFFu) << 16));          // tensor_dim1 lo16
  g1[3] = (int)((((unsigned)NC >> 16) & 0xFFFFu)            // tensor_dim1 hi16
              | (((unsigned)TT) << 16));                    // tile_dim0 = 64
  g1[4] = (int)NC;                                          // tile_dim1=224, tile_dim2=0
  g1[5] = (int)NT;                                          // tensor_dim0_stride lo32
  g1[6] = 0;                                                // stride hi | dim1_stride lo
  g1[7] = 0;
  i32x4 z4 = {0, 0, 0, 0};                                  // groups 2/3 unused (2-D)
  i32x8 z8 = {0, 0, 0, 0, 0, 0, 0, 0};                      // unused extra operand
  __builtin_amdgcn_tensor_load_to_lds(g0, g1, z4, z4, z8, 0);
}

__global__ __launch_bounds__(256) void agss_fused(const float* __restrict__ x,
                                                  const float* __restrict__ lw,
                                                  float* __restrict__ out) {
  __shared__ float xs[2][NC * STR];  // double-buffered x tile (2 x 58.2 KB)
  __shared__ float is_[NC * STR];    // I, then spikes (58.2 KB)
  __shared__ float lwsh[NC];
  __shared__ float wsh[TT];          // raw sigma dot per t
  __shared__ float ker[TT * 16];     // 13 taps, stride 16

  const int tid = threadIdx.x;
  const int b = blockIdx.x;
  const float* xb = x + (size_t)b * NC * NT;
  float* ob = out + (size_t)b * NC * NT;
  const int wv = tid >> 5;  // wave index (wave-uniform)

  for (int i = tid; i < NC; i += 256) lwsh[i] = lw[i];

  // Prologue: async-DMA tile 0 into buffer 0 (wave 0 only; EXEC-independent,
  // one TENSOR op per wave -> guard to avoid duplicate DMAs).
  if (wv == 0) tdm_load_tile(xb, lds_off(&xs[0][0]));

  float mem = 0.0f;  // LIF state: thread c (<224) owns channel c across tiles

  for (int tile = 0; tile < NTILES; ++tile) {
    const int t0 = tile * TT;
    const int cur = tile & 1;
    const float* buf = &xs[cur][0];

    // ---- 0) tile data ready: drain TENSORcnt on the issuing wave, publish ----
    if (wv == 0) __builtin_amdgcn_s_wait_tensorcnt(0);
    __syncthreads();
    // Prefetch next tile into the other buffer while we compute this one.
    // (That buffer's last DS reads completed before each wave's barrier above.)
    if (wv == 0 && tile + 1 < NTILES)
      tdm_load_tile(xb + (size_t)(tile + 1) * TT, lds_off(&xs[cur ^ 1][0]));

    // ---- 2) sigma dot over channels via V_WMMA_F32_16X16X4_F32 ----
    // waves 0..3 each produce 16 timesteps; B columns replicate lw so every
    // column of D equals the length-224 dot product.
    if (wv < 4) {
      int lane = tid & 31;
      int m = lane & 15;            // A row / D column index
      int kb = (lane < 16) ? 0 : 2; // K pair owned by this half-wave
      int col = wv * 16 + m;        // timestep within tile
      v8f acc = {0.f, 0.f, 0.f, 0.f, 0.f, 0.f, 0.f, 0.f};
      for (int s = 0; s < NC / 4; ++s) {
        int cb = 4 * s + kb;
        v2f a, bm;
        a.x = buf[cb * STR + col];         // A[m][kb]
        a.y = buf[(cb + 1) * STR + col];   // A[m][kb+1]
        bm.x = lwsh[cb];                   // B[kb][*]
        bm.y = lwsh[cb + 1];               // B[kb+1][*]
        acc = __builtin_amdgcn_wmma_f32_16x16x4_f32(
            false, a, false, bm, (short)0, acc, false, false);
      }
      if (m == 0) {                 // lanes N==0 hold D rows 0-7 / 8-15
        int mo = (lane < 16) ? 0 : 8;
        #pragma unroll
        for (int p = 0; p < 8; ++p) wsh[wv * 16 + mo + p] = acc[p];
      }
    }
    __syncthreads();

    // ---- 3) per-timestep Gaussian kernel + 130-point normalizer ----
    if (tid < TT) {
      float sig = F_SIGMA_MEAN + (F_SIGMA_MAX - F_SIGMA_MIN) * wsh[tid];
      sig = fminf(fmaxf(sig, F_SIGMA_MIN), F_SIGMA_MAX);
      float qi = -0.5f / (sig * sig);
      float norm = 0.0f;
      for (int mm = 0; mm < 130; ++mm) {
        float rr = -60.0f + (float)mm * (120.0f / 129.0f);
        norm += expf(rr * rr * qi);
      }
      float rnorm = 1.0f / norm;
      #pragma unroll
      for (int k = 0; k < 13; ++k) {
        float rr = (float)(k - 6);
        ker[tid * 16 + k] = expf(rr * rr * qi) * rnorm;
      }
    }
    __syncthreads();

    // ---- 4) 13-tap channel smoothing, I = min(x, x_mean) ----
    for (int idx = tid; idx < NC * TT; idx += 256) {
      int c = idx % NC;   // lanes contiguous in c -> conflict-free (stride 65)
      int tt = idx / NC;  // uniform per 32-lane block (224 = 7*32)
      float xm = 0.0f;
      #pragma unroll
      for (int k = 0; k < 13; ++k) {
        int cc = c + k - 6;
        float xv = (cc >= 0 && cc < NC) ? buf[cc * STR + tt] : 0.0f;
        xm = fmaf(xv, ker[tt * 16 + k], xm);
      }
      float xv = buf[c * STR + tt];
      is_[c * STR + tt] = fminf(xv, xm);  // x - relu(x - x_mean) == min
    }
    __syncthreads();

    // ---- 5) LIF scan over this tile; spikes rewritten in place ----
    if (tid < NC) {
      float* row = is_ + tid * STR;
      for (int tt = 0; tt < TT; ++tt) {
        float It = row[tt];
        float reset = (mem > F_THRESH) ? F_THRESH : 0.0f;
        mem = F_BETA * mem + It - reset;
        row[tt] = (mem > F_THRESH) ? 1.0f : 0.0f;
      }
    }
    __syncthreads();

    // ---- 6) coalesced spike writeback ----
    for (int idx = tid; idx < NC * TT; idx += 256) {
      int tt = idx & (TT - 1);
      int c = idx >> 6;
      ob[(size_t)c * NT + t0 + tt] = is_[c * STR + tt];
    }
    // no trailing barrier needed: next iteration's wait+barrier republishes,
    // and is_/xs are only overwritten after those barriers.
  }
}

extern "C" void kernel_launch(void* const* d_in, const int* in_sizes, int n_in,
                              void* d_out, int out_size, void* d_ws, size_t ws_size,
                              hipStream_t stream) {
  const float* x = (const float*)d_in[0];    // (32,1,224,2048) f32
  const float* lw = (const float*)d_in[1];   // (1,224) f32
  float* out = (float*)d_out;                // (32,1,224,2048) f32
  agss_fused<<<NB, 256, 0, stream>>>(x, lw, out);
}